// GCN_34677565948890
// MI455X (gfx1250) — compile-verified
//
#include <hip/hip_runtime.h>

typedef float v2f __attribute__((ext_vector_type(2)));
typedef float v8f __attribute__((ext_vector_type(8)));

#define N_NODES 100000
#define FEAT_IN 128
#define HID     16
#define OUTC    7
#define OPAD    8
#define NTILES  (N_NODES / 16)   // 6250 exactly

// workspace layout in float units
#define OFF_DINV 0
#define OFF_H1   131072                       // 512KB-aligned
#define OFF_AGG1 (OFF_H1   + N_NODES * HID)   // 1731072
#define OFF_H2   (OFF_AGG1 + N_NODES * HID)   // 3331072
#define OFF_AGG2 (OFF_H2   + N_NODES * OPAD)  // 4131072

// ---- degree / normalization ---------------------------------------------
__global__ void k_deg_init(float* __restrict__ deg) {
    int i = blockIdx.x * 256 + threadIdx.x;
    if (i < N_NODES) deg[i] = 1.0f;           // self-loop contributes 1
}

__global__ void k_deg_count(const long long* __restrict__ dst, int E,
                            float* __restrict__ deg) {
    int e = blockIdx.x * 256 + threadIdx.x;
    if (e < E) atomicAdd(&deg[(int)dst[e]], 1.0f);
}

__global__ void k_dinv(float* __restrict__ deg) {
    int i = blockIdx.x * 256 + threadIdx.x;
    if (i < N_NODES) deg[i] = rsqrtf(deg[i]); // deg >= 1 always
}

// ---- GEMM1: h1[100000x16] = x[100000x128] @ W1[128x16]  (f32 WMMA) ------
__global__ void __launch_bounds__(256)
k_gemm1(const float* __restrict__ x, const float* __restrict__ W1,
        float* __restrict__ h1) {
    const int tile = (blockIdx.x * 256 + (int)threadIdx.x) >> 5; // wave id
    if (tile >= NTILES) return;                                  // wave-uniform
    const int lane = threadIdx.x & 31;
    const int half = lane >> 4;       // 0: K={k,k+1}, 1: K={k+2,k+3}
    const int l15  = lane & 15;
    const int koff = half * 2;
    const float* xrow = x + (size_t)(tile * 16 + l15) * FEAT_IN;

    v8f c = {};
    #pragma unroll 4
    for (int k = 0; k < FEAT_IN; k += 4) {
        v2f a = *(const v2f*)(xrow + k + koff);          // A 16x4 fragment
        v2f b;
        b.x = W1[(k + koff    ) * HID + l15];            // B 4x16 fragment
        b.y = W1[(k + koff + 1) * HID + l15];
        c = __builtin_amdgcn_wmma_f32_16x16x4_f32(false, a, false, b,
                                                  (short)0, c, false, false);
    }
    #pragma unroll
    for (int j = 0; j < 8; ++j)
        h1[(size_t)(tile * 16 + j + 8 * half) * HID + l15] = c[j];
}

// ---- init agg1 with self-loop term + bias --------------------------------
__global__ void k_init_agg1(const float* __restrict__ h1,
                            const float* __restrict__ dinv,
                            const float* __restrict__ b1,
                            float* __restrict__ agg1) {
    int idx = blockIdx.x * 256 + threadIdx.x;
    if (idx >= N_NODES * HID) return;
    int i = idx >> 4, f = idx & 15;
    float di = dinv[i];
    agg1[idx] = h1[idx] * di * di + b1[f];
}

// ---- layer-1 edge scatter: 16 lanes per edge -----------------------------
__global__ void k_edge1(const long long* __restrict__ src,
                        const long long* __restrict__ dst, int E,
                        const float* __restrict__ h1,
                        const float* __restrict__ dinv,
                        float* __restrict__ agg1) {
    int id = blockIdx.x * 256 + threadIdx.x;
    int e = id >> 4;
    if (e >= E) return;
    int f = id & 15;
    int s = (int)src[e], d = (int)dst[e];
    float norm = dinv[s] * dinv[d];
    atomicAdd(&agg1[d * HID + f], h1[s * HID + f] * norm);
}

// ---- GEMM2: h2 = relu(agg1) @ pad16(W2[16x7])  (f32 WMMA, K=16) ----------
__global__ void __launch_bounds__(256)
k_gemm2(const float* __restrict__ agg1, const float* __restrict__ W2,
        float* __restrict__ h2) {
    const int tile = (blockIdx.x * 256 + (int)threadIdx.x) >> 5;
    if (tile >= NTILES) return;                 // wave-uniform
    const int lane = threadIdx.x & 31;
    const int half = lane >> 4;
    const int l15  = lane & 15;
    const int koff = half * 2;
    const float* arow = agg1 + (size_t)(tile * 16 + l15) * HID;

    v8f c = {};
    #pragma unroll
    for (int k = 0; k < HID; k += 4) {
        v2f raw = *(const v2f*)(arow + k + koff);
        v2f a;
        a.x = fmaxf(raw.x, 0.0f);               // fused ReLU
        a.y = fmaxf(raw.y, 0.0f);
        v2f b;
        b.x = (l15 < OUTC) ? W2[(k + koff    ) * OUTC + l15] : 0.0f;
        b.y = (l15 < OUTC) ? W2[(k + koff + 1) * OUTC + l15] : 0.0f;
        c = __builtin_amdgcn_wmma_f32_16x16x4_f32(false, a, false, b,
                                                  (short)0, c, false, false);
    }
    if (l15 < OUTC) {
        #pragma unroll
        for (int j = 0; j < 8; ++j)
            h2[(size_t)(tile * 16 + j + 8 * half) * OPAD + l15] = c[j];
    }
}

// ---- init agg2 with self-loop term + bias --------------------------------
__global__ void k_init_agg2(const float* __restrict__ h2,
                            const float* __restrict__ dinv,
                            const float* __restrict__ b2,
                            float* __restrict__ agg2) {
    int idx = blockIdx.x * 256 + threadIdx.x;
    if (idx >= N_NODES * OPAD) return;
    int i = idx >> 3, f = idx & 7;
    if (f < OUTC) {
        float di = dinv[i];
        agg2[idx] = h2[idx] * di * di + b2[f];
    }
}

// ---- layer-2 edge scatter: 8 lanes per edge (7 active) -------------------
__global__ void k_edge2(const long long* __restrict__ src,
                        const long long* __restrict__ dst, int E,
                        const float* __restrict__ h2,
                        const float* __restrict__ dinv,
                        float* __restrict__ agg2) {
    int id = blockIdx.x * 256 + threadIdx.x;
    int e = id >> 3;
    if (e >= E) return;
    int f = id & 7;
    if (f >= OUTC) return;
    int s = (int)src[e], d = (int)dst[e];
    float norm = dinv[s] * dinv[d];
    atomicAdd(&agg2[d * OPAD + f], h2[s * OPAD + f] * norm);
}

// ---- log_softmax over 7 classes ------------------------------------------
__global__ void k_logsoftmax(const float* __restrict__ agg2,
                             float* __restrict__ out) {
    int i = blockIdx.x * 256 + threadIdx.x;
    if (i >= N_NODES) return;
    float v[OUTC], m = -3.4e38f;
    #pragma unroll
    for (int j = 0; j < OUTC; ++j) { v[j] = agg2[i * OPAD + j]; m = fmaxf(m, v[j]); }
    float s = 0.0f;
    #pragma unroll
    for (int j = 0; j < OUTC; ++j) s += expf(v[j] - m);
    float lse = logf(s);
    #pragma unroll
    for (int j = 0; j < OUTC; ++j) out[i * OUTC + j] = v[j] - m - lse;
}

extern "C" void kernel_launch(void* const* d_in, const int* in_sizes, int n_in,
                              void* d_out, int out_size, void* d_ws, size_t ws_size,
                              hipStream_t stream) {
    (void)n_in; (void)out_size; (void)ws_size;
    const float*     x  = (const float*)d_in[0];
    const long long* ei = (const long long*)d_in[1];   // int64 [2, E]
    const float*     W1 = (const float*)d_in[2];
    const float*     b1 = (const float*)d_in[3];
    const float*     W2 = (const float*)d_in[4];
    const float*     b2 = (const float*)d_in[5];
    const int E = in_sizes[1] / 2;
    const long long* src = ei;
    const long long* dst = ei + E;

    float* ws   = (float*)d_ws;
    float* dinv = ws + OFF_DINV;
    float* h1   = ws + OFF_H1;
    float* agg1 = ws + OFF_AGG1;
    float* h2   = ws + OFF_H2;
    float* agg2 = ws + OFF_AGG2;

    const int nb_nodes = (N_NODES + 255) / 256;

    k_deg_init   <<<nb_nodes, 256, 0, stream>>>(dinv);
    k_deg_count  <<<(E + 255) / 256, 256, 0, stream>>>(dst, E, dinv);
    k_dinv       <<<nb_nodes, 256, 0, stream>>>(dinv);

    k_gemm1      <<<(NTILES + 7) / 8, 256, 0, stream>>>(x, W1, h1);
    k_init_agg1  <<<(N_NODES * HID + 255) / 256, 256, 0, stream>>>(h1, dinv, b1, agg1);
    k_edge1      <<<(E * 16 + 255) / 256, 256, 0, stream>>>(src, dst, E, h1, dinv, agg1);

    k_gemm2      <<<(NTILES + 7) / 8, 256, 0, stream>>>(agg1, W2, h2);
    k_init_agg2  <<<(N_NODES * OPAD + 255) / 256, 256, 0, stream>>>(h2, dinv, b2, agg2);
    k_edge2      <<<(E * 8 + 255) / 256, 256, 0, stream>>>(src, dst, E, h2, dinv, agg2);

    k_logsoftmax <<<nb_nodes, 256, 0, stream>>>(agg2, (float*)d_out);
}